// GCN_85684597555226
// MI455X (gfx1250) — compile-verified
//
#include <hip/hip_runtime.h>

#define GCN_N_NODES 100000
#define GCN_N_EDGES 1600000

typedef __attribute__((ext_vector_type(2))) float v2f;
typedef __attribute__((ext_vector_type(8))) float v8f;

// ---------------------------------------------------------------------------
// Degrees via float atomics (counts -> later transformed to rsqrt norms).
// ---------------------------------------------------------------------------
__global__ void gcn_degree_kernel(const int* __restrict__ src, const int* __restrict__ dst,
                                  float* __restrict__ degOut, float* __restrict__ degIn, int nE) {
  int e = blockIdx.x * blockDim.x + threadIdx.x;
  if (e < nE) {
    atomicAdd(&degOut[src[e]], 1.0f);
    atomicAdd(&degIn[dst[e]], 1.0f);
  }
}

// In-place: deg -> max(deg,1)^-0.5  (both deg arrays are contiguous)
__global__ void gcn_norm_kernel(float* __restrict__ deg, int n) {
  int i = blockIdx.x * blockDim.x + threadIdx.x;
  if (i < n) deg[i] = rsqrtf(fmaxf(deg[i], 1.0f));
}

// ---------------------------------------------------------------------------
// SpMM aggregate: agg[dst] += h[src] * norm_out[src].
// 16 threads per edge, float4 gather + 4 global f32 atomics each.
// Node features (25.6MB) are resident in the 192MB L2, so this is
// L2-atomic-throughput bound, not HBM bound. Pure 32-bit indexing.
// ---------------------------------------------------------------------------
__global__ void gcn_aggregate_kernel(const float* __restrict__ h, const float* __restrict__ normOut,
                                     const int* __restrict__ src, const int* __restrict__ dst,
                                     float* __restrict__ agg, int nE) {
  unsigned i = blockIdx.x * blockDim.x + threadIdx.x;        // < 25.6M, fits u32
  if (i >= 16u * (unsigned)nE) return;
  int e = (int)(i >> 4);
  int f = ((int)i & 15) << 2;
  int s = src[e];
  int d = dst[e];
  float no = normOut[s];
  float4 v = *(const float4*)(h + s * 64 + f);
  float* a = agg + d * 64 + f;
  atomicAdd(a + 0, v.x * no);
  atomicAdd(a + 1, v.y * no);
  atomicAdd(a + 2, v.z * no);
  atomicAdd(a + 3, v.w * no);
}

// ---------------------------------------------------------------------------
// Dense tail: Out = act((norm_in[:,None]*Agg) @ W + b), fp32 WMMA 16x16x4.
// One wave per 16x16 output tile, K=64 as 16 WMMA steps.
//
// W (64 x NCOLS, <=16KB) is staged once per block into LDS, K-pair
// interleaved:  Wlds[(r>>1)*LSTRIDE + 2*c + (r&1)] = W[r][c]
// so each step's B fragment is one 8-byte ds_load_b64. LSTRIDE = 2*NCOLS+2
// keeps the two half-wave K-rows on disjoint LDS banks.
//
// A 16x4 layout : lanes 0-15 -> M=0..15 (K=0,1), lanes 16-31 -> K=2,3
// B 4x16 layout : lanes 0-15 -> N=0..15 (K=0,1), lanes 16-31 -> K=2,3
// C/D 16x16     : VGPR v holds M = v + 8*(lane>=16), N = lane&15
// ---------------------------------------------------------------------------
template <int NCOLS, bool RELU>
__global__ void gcn_gemm_wmma_kernel(const float* __restrict__ Agg, const float* __restrict__ normIn,
                                     const float* __restrict__ W, const float* __restrict__ bias,
                                     float* __restrict__ Out, int M, int tilesN) {
  constexpr int LSTRIDE = 2 * NCOLS + 2;
  __shared__ float Wlds[32 * LSTRIDE];

  // cooperative stage of W into LDS (interleaved K-pairs)
  for (int idx = threadIdx.x; idx < 64 * NCOLS; idx += blockDim.x) {
    int r = idx / NCOLS;            // K row 0..63  (NCOLS is constexpr -> shift)
    int c = idx - r * NCOLS;        // column
    Wlds[(r >> 1) * LSTRIDE + (c << 1) + (r & 1)] = W[idx];
  }
  __syncthreads();

  int wavesPerBlock = blockDim.x >> 5;
  int tile = blockIdx.x * wavesPerBlock + ((int)threadIdx.x >> 5);
  int tilesM = M >> 4;
  if (tile >= tilesM * tilesN) return;          // wave-uniform (grids sized exactly)
  int tM = tile / tilesN;
  int tN = tile - tM * tilesN;

  int lane = (int)threadIdx.x & 31;
  int mn   = lane & 15;                         // A-row within tile / B-col within tile
  int half = lane >> 4;                         // selects K half-pair
  int row  = (tM << 4) + mn;
  int col  = (tN << 4) + mn;

  float nin = normIn[row];
  const float* Arow = Agg + row * 64;           // 32-bit index math only
  const int bbase = (col << 1);                 // LDS float offset of (pair, col)

  v8f acc = {};
#pragma unroll
  for (int k = 0; k < 64; k += 4) {
    v2f a = *(const v2f*)(Arow + k + (half << 1));        // 8B aligned
    a *= nin;                                             // fold D_in^{-1/2} into A
    int pidx = (k >> 1) + half;                           // K-pair index 0..31
    v2f b = *(const v2f*)(&Wlds[pidx * LSTRIDE + bbase]); // ds_load_b64
    acc = __builtin_amdgcn_wmma_f32_16x16x4_f32(
        /*neg_a=*/false, a, /*neg_b=*/false, b,
        /*c_mod=*/(short)0, acc, /*reuse_a=*/false, /*reuse_b=*/false);
  }

  float bv = bias[col];
  int obase = ((tM << 4) + (half << 3)) * NCOLS + col;    // 32-bit
#pragma unroll
  for (int v = 0; v < 8; ++v) {
    float val = acc[v] + bv;
    if (RELU) val = fmaxf(val, 0.0f);
    Out[obase + v * NCOLS] = val;
  }
}

// ---------------------------------------------------------------------------
// Host-side orchestration (graph-capture safe: only async ops on `stream`).
// ---------------------------------------------------------------------------
extern "C" void kernel_launch(void* const* d_in, const int* in_sizes, int n_in,
                              void* d_out, int out_size, void* d_ws, size_t ws_size,
                              hipStream_t stream) {
  (void)in_sizes; (void)n_in; (void)out_size; (void)ws_size;

  const float* x   = (const float*)d_in[0];
  const int*   src = (const int*)d_in[1];
  const int*   dst = (const int*)d_in[2];
  const float* W1  = (const float*)d_in[3];
  const float* b1  = (const float*)d_in[4];
  const float* W2  = (const float*)d_in[5];
  const float* b2  = (const float*)d_in[6];
  const float* W3  = (const float*)d_in[7];
  const float* b3  = (const float*)d_in[8];
  float* out = (float*)d_out;

  float* ws      = (float*)d_ws;
  float* normOut = ws;                                  // N
  float* normIn  = ws + GCN_N_NODES;                    // N
  float* agg     = ws + 2 * (size_t)GCN_N_NODES;        // N*64
  float* hbuf    = agg + (size_t)GCN_N_NODES * 64;      // N*64
  // total workspace: 130 * N * 4B = 52 MB

  // --- degrees -> norms ---
  hipMemsetAsync(normOut, 0, 2 * (size_t)GCN_N_NODES * sizeof(float), stream);
  gcn_degree_kernel<<<(GCN_N_EDGES + 255) / 256, 256, 0, stream>>>(src, dst, normOut, normIn, GCN_N_EDGES);
  gcn_norm_kernel<<<(2 * GCN_N_NODES + 255) / 256, 256, 0, stream>>>(normOut, 2 * GCN_N_NODES);

  const long long aggThreads = (long long)GCN_N_EDGES * 16;
  const int aggBlocks = (int)((aggThreads + 255) / 256);

  // Exact grids: no partially-active blocks (barrier-safe, EXEC all-1s).
  const int tiles64 = (GCN_N_NODES / 16) * 4;           // 25000 tiles
  const int tiles32 = (GCN_N_NODES / 16) * 2;           // 12500 tiles
  const int gemmBlocks64 = tiles64 / 8;                 // 256 thr = 8 waves -> 3125
  const int gemmBlocks32 = tiles32 / 4;                 // 128 thr = 4 waves -> 3125

  // --- layer 1: x -> hbuf ---
  hipMemsetAsync(agg, 0, (size_t)GCN_N_NODES * 64 * sizeof(float), stream);
  gcn_aggregate_kernel<<<aggBlocks, 256, 0, stream>>>(x, normOut, src, dst, agg, GCN_N_EDGES);
  gcn_gemm_wmma_kernel<64, true><<<gemmBlocks64, 256, 0, stream>>>(agg, normIn, W1, b1, hbuf, GCN_N_NODES, 4);

  // --- layer 2: hbuf -> hbuf ---
  hipMemsetAsync(agg, 0, (size_t)GCN_N_NODES * 64 * sizeof(float), stream);
  gcn_aggregate_kernel<<<aggBlocks, 256, 0, stream>>>(hbuf, normOut, src, dst, agg, GCN_N_EDGES);
  gcn_gemm_wmma_kernel<64, true><<<gemmBlocks64, 256, 0, stream>>>(agg, normIn, W2, b2, hbuf, GCN_N_NODES, 4);

  // --- layer 3: hbuf -> out (32 classes, no relu) ---
  hipMemsetAsync(agg, 0, (size_t)GCN_N_NODES * 64 * sizeof(float), stream);
  gcn_aggregate_kernel<<<aggBlocks, 256, 0, stream>>>(hbuf, normOut, src, dst, agg, GCN_N_EDGES);
  gcn_gemm_wmma_kernel<32, false><<<gemmBlocks32, 128, 0, stream>>>(agg, normIn, W3, b3, out, GCN_N_NODES, 2);
}